// GaussianBlur2d_84000970375243
// MI455X (gfx1250) — compile-verified
//
#include <hip/hip_runtime.h>
#include <hip/hip_bf16.h>

// Gaussian blur 13x13, sigma=2, reflect pad, 32x1x1024x1024 fp32.
// Separable single-pass (H then V through LDS), math on the matrix pipe via
// V_WMMA_F32_16X16X4_F32 (7 chained WMMAs = K=28 banded GEMM per 16x16 tile).
// Min HBM traffic: 1 read + 1 write of the image (~268 MB -> ~11.5us floor).

typedef float v2f __attribute__((ext_vector_type(2)));
typedef float v8f __attribute__((ext_vector_type(8)));

#define IMG_H 1024
#define IMG_W 1024
#define TILE 64            // output tile per workgroup (64x64)
#define HALO 6             // 13-tap -> +/-6
#define XROWS 80           // 76 needed, rounded to 5 blocks of 16
#define XSTRIDE 76         // 76 cols needed; stride 76 is bank-friendly (see notes)

__device__ __forceinline__ int refl(int i) {
    if (i < 0) i = -i;
    if (i >= IMG_H) i = 2 * IMG_H - 2 - i;   // 2046 - i
    return i;
}

__global__ __launch_bounds__(256)
void GaussianBlur2d_84000970375243_kernel(const float* __restrict__ x,
                                          const float* __restrict__ kern2d,
                                          float* __restrict__ out) {
    __shared__ float gk[16];                 // normalized 1D gaussian, 13 taps
    __shared__ float Xs[XROWS * XSTRIDE];    // input tile + halo (rows 76..79 zero)
    __shared__ float Ts[XROWS * XSTRIDE];    // horizontally blurred intermediate

    const int tid  = threadIdx.x;
    const int lane = tid & 31;
    const int n16  = lane & 15;              // M (A/C) or N (B/C) per WMMA layout
    const int h    = (lane >> 4) & 1;        // lane half selects K group
    const int wave = tid >> 5;               // 8 waves

    const int x0 = blockIdx.x * TILE;
    const int y0 = blockIdx.y * TILE;
    const size_t plane = (size_t)blockIdx.z * IMG_H * IMG_W;
    const float* __restrict__ xin = x + plane;
    float* __restrict__ op = out + plane;

    // --- 1D taps from the supplied 2D kernel: k = gn*gn^T  =>  gn[j] = k[6][j]/sqrt(k[6][6])
    if (tid == 0) {
        float s = sqrtf(kern2d[6 * 13 + 6]);
        #pragma unroll
        for (int j = 0; j < 13; ++j) gk[j] = kern2d[6 * 13 + j] / s;
        gk[13] = 0.f; gk[14] = 0.f; gk[15] = 0.f;
    }

    // --- Load input tile (with reflect halo) into LDS; zero the 4 pad rows.
    for (int idx = tid; idx < XROWS * XSTRIDE; idx += 256) {
        int r = idx / XSTRIDE;
        int c = idx - r * XSTRIDE;
        float v = 0.f;
        if (r < 2 * HALO + TILE) {           // 76 real rows
            int gr = refl(y0 - HALO + r);
            int gc = refl(x0 - HALO + c);
            v = xin[(size_t)gr * IMG_W + gc];
        }
        Xs[idx] = v;
    }
    __syncthreads();

    // --- Banded-weight registers. W[m,r] = g[r-m]; per WMMA 32-bit layout both
    // the A operand (vertical pass) and B operand (horizontal pass) need
    // element K=2h+j for chunk c:  wr[c][j] = g[4c + 2h + j - n16].
    v2f wr[7];
    #pragma unroll
    for (int c = 0; c < 7; ++c) {
        #pragma unroll
        for (int j = 0; j < 2; ++j) {
            int idx = 4 * c + 2 * h + j - n16;
            wr[c][j] = (idx >= 0 && idx < 13) ? gk[idx] : 0.f;
        }
    }

    // --- Horizontal pass: T(16x16 tile) = X(16x28) * W^T(28x16), 7 WMMAs.
    // 5 row-blocks x 4 col-blocks = 20 tiles over 8 waves.
    for (int t = wave; t < 20; t += 8) {
        int rb = t >> 2, cb = t & 3;
        int m0 = rb * 16, n0 = cb * 16;
        v8f acc = {};
        #pragma unroll
        for (int c = 0; c < 7; ++c) {
            const float* row = &Xs[(m0 + n16) * XSTRIDE + n0 + 4 * c + 2 * h];
            v2f a = { row[0], row[1] };      // A: a[j] = X[m0+M, n0+4c+2h+j]
            acc = __builtin_amdgcn_wmma_f32_16x16x4_f32(
                false, a, false, wr[c], (short)0, acc, false, false);
        }
        #pragma unroll
        for (int j = 0; j < 8; ++j)          // C/D: vgpr j -> row M = j + 8h
            Ts[(m0 + j + 8 * h) * XSTRIDE + n0 + n16] = acc[j];
    }
    __syncthreads();

    // --- Vertical pass: O(16x16) = W(16x28) * T(28x16), 7 WMMAs.
    // 4x4 = 16 tiles over 8 waves.
    for (int t = wave; t < 16; t += 8) {
        int rb = t >> 2, cb = t & 3;
        int m0 = rb * 16, n0 = cb * 16;
        v8f acc = {};
        #pragma unroll
        for (int c = 0; c < 7; ++c) {
            int r = m0 + 4 * c + 2 * h;      // B: b[v] = T[m0+4c+2h+v, n0+N]
            v2f b = { Ts[r * XSTRIDE + n0 + n16],
                      Ts[(r + 1) * XSTRIDE + n0 + n16] };
            acc = __builtin_amdgcn_wmma_f32_16x16x4_f32(
                false, wr[c], false, b, (short)0, acc, false, false);
        }
        #pragma unroll
        for (int j = 0; j < 8; ++j)
            op[(size_t)(y0 + m0 + j + 8 * h) * IMG_W + x0 + n0 + n16] = acc[j];
    }
}

extern "C" void kernel_launch(void* const* d_in, const int* in_sizes, int n_in,
                              void* d_out, int out_size, void* d_ws, size_t ws_size,
                              hipStream_t stream) {
    const float* x    = (const float*)d_in[0];   // 32x1x1024x1024 fp32
    const float* kern = (const float*)d_in[1];   // 1x1x13x13 fp32
    float* out = (float*)d_out;                  // 32x1x1024x1024 fp32
    (void)in_sizes; (void)n_in; (void)out_size; (void)d_ws; (void)ws_size;

    dim3 grid(IMG_W / TILE, IMG_H / TILE, 32);   // 16 x 16 x 32
    dim3 block(256);                              // 8 waves (wave32)
    GaussianBlur2d_84000970375243_kernel<<<grid, block, 0, stream>>>(x, kern, out);
}